// TPFirst_18511309046142
// MI455X (gfx1250) — compile-verified
//
#include <hip/hip_runtime.h>

typedef float v2f __attribute__((ext_vector_type(2)));
typedef float v8f __attribute__((ext_vector_type(8)));

#define NS   48
#define NV   10
#define WPZ  (NS*NS + NS*NV)   // 2784 weights per sample
#define OUTW (NS + 3*NV)       // 78 outputs per sample

// NORM_0E = sqrt(1/48); NORM_1O = sqrt(3/48)/sqrt(3) = sqrt(1/48) (identical)
#define NORM_0E 0.14433756729740643f
#define NORM_1O 0.14433756729740643f

// One wave (32 lanes) handles one sample z.
// A-matrix rows are all replicated with the x1 row, so every row of the WMMA
// D-matrix equals y = x1 . W; accumulator VGPR0 then holds y[w0 + (lane&15)]
// in every lane (rows M=0 and M=8 are identical).
//
// All weight loads are immediate-offset loads from two per-lane base pointers,
// so the unrolled K-loop has no per-iteration address arithmetic: just
// s_clause'd global_load_b32 runs + v_wmma.
__global__ __launch_bounds__(256) void tp_first_wmma_kernel(
    const float* __restrict__ x1,
    const float* __restrict__ x2,
    const float* __restrict__ weights,
    float* __restrict__ out,
    int Z)
{
    const int lane = threadIdx.x & 31;
    const int wave = threadIdx.x >> 5;
    const int z    = blockIdx.x * 8 + wave;
    if (z >= Z) return;               // whole wave exits together: EXEC stays all-1s

    const int half = lane >> 4;       // 0: lanes 0-15, 1: lanes 16-31
    const int col  = lane & 15;       // N index within a 16-wide tile
    const int colw = (col < NV) ? col : (NV - 1);   // clamp w01 column (stay in-bounds)

    const float* w00 = weights + (size_t)z * WPZ;   // [u*48 + w]
    const float* x2z = x2 + (size_t)z * 9;

    // Per-lane base pointers; every load below is a constant IOFFSET from one
    // of these (after full unroll).
    const float* pA  = x1 + (size_t)z * NS + 2 * half;        // A: +ks*4
    const float* pB0 = w00 + 2 * half * NS + col;             // w00: +ks*192 + {0,16,32} + {0,48}
    const float* pB1 = w00 + NS * NS + 2 * half * NV + colw;  // w01: +ks*40 + {0,10}

    v8f acc0 = {}; v8f acc1 = {}; v8f acc2 = {}; v8f acc3 = {};

#pragma unroll
    for (int ks = 0; ks < 12; ++ks) {
        // A: every lane of a half loads the same x1 pair -> all 16 A rows = x1 row
        v2f a = *(const v2f*)(pA + ks * 4);

        // B tiles: lane 'col' supplies column w0+col of W for K=u0 (x) and K=u0+1 (y)
        const int o = ks * 4 * NS;    // = ks*192, compile-time constant per iteration
        const int q = ks * 4 * NV;    // = ks*40
        v2f b0 = { pB0[o +  0], pB0[o + NS +  0] };
        v2f b1 = { pB0[o + 16], pB0[o + NS + 16] };
        v2f b2 = { pB0[o + 32], pB0[o + NS + 32] };
        v2f b3 = { pB1[q +  0], pB1[q + NV +  0] };

        acc0 = __builtin_amdgcn_wmma_f32_16x16x4_f32(false, a, false, b0, (short)0, acc0, false, false);
        acc1 = __builtin_amdgcn_wmma_f32_16x16x4_f32(false, a, false, b1, (short)0, acc1, false, false);
        acc2 = __builtin_amdgcn_wmma_f32_16x16x4_f32(false, a, false, b2, (short)0, acc2, false, false);
        acc3 = __builtin_amdgcn_wmma_f32_16x16x4_f32(false, a, false, b3, (short)0, acc3, false, false);
    }

    float* outz = out + (size_t)z * OUTW;

    // ---- result_0e: out[w] = NORM_0E * x2[z,0] * y0[w] ----
    const float s0 = NORM_0E * x2z[0];
    // w = 0..31: lanes 0-15 take tile0, lanes 16-31 take tile1 -> contiguous 128B store
    const float v01 = s0 * (half ? acc1[0] : acc0[0]);
    outz[lane] = v01;
    // w = 32..47: lanes 0-15 store tile2
    const float v2 = s0 * acc2[0];
    if (half == 0) outz[32 + col] = v2;

    // ---- result_1o: out[48 + w*3 + j] = NORM_1O * x2[z,1+j] * y1[w] ----
    const int wi = lane / 3;          // 0..10 (only lanes < 30 store)
    const int ji = lane - 3 * wi;     // 0..2
    // fan y1[w] out from lane w (w < 10, all source lanes active)
    const float y1w = __shfl(acc3[0], (wi < NV) ? wi : 0, 32);
    const float x2j = x2z[1 + ji];
    if (lane < 3 * NV) outz[NS + lane] = NORM_1O * x2j * y1w;
}

extern "C" void kernel_launch(void* const* d_in, const int* in_sizes, int n_in,
                              void* d_out, int out_size, void* d_ws, size_t ws_size,
                              hipStream_t stream) {
    const float* x1      = (const float*)d_in[0];
    const float* x2      = (const float*)d_in[1];
    const float* weights = (const float*)d_in[2];
    float* out = (float*)d_out;

    const int Z = in_sizes[0] / NS;          // x1 is (Z, 48)
    const int waves_per_block = 8;           // 256 threads = 8 wave32
    const int blocks = (Z + waves_per_block - 1) / waves_per_block;

    tp_first_wmma_kernel<<<blocks, 256, 0, stream>>>(x1, x2, weights, out, Z);
}